// FastMultiEMAMixture_6433861009533
// MI455X (gfx1250) — compile-verified
//
#include <hip/hip_runtime.h>
#include <cstdint>

// Problem constants (from the reference)
#define BB   256
#define TL   336
#define CC   512
#define KK   3

// Tiling
#define VEC       2              // channels per thread
#define NTHREADS  (CC / VEC)     // 256 threads = one batch's full channel dim
#define TT        16             // time steps per async stage
#define NSTAGE    (TL / TT)      // 21 stages, exact
#define NBUF      3              // triple buffer, prefetch distance 2

// Native clang vector type (HIP's float2 class is rejected by
// __builtin_nontemporal_store).
typedef float vfloat2 __attribute__((ext_vector_type(2)));

// 16 async b64 global->LDS issues; the signed 24-bit instruction offset advances
// BOTH the LDS destination and the global source by CC*4 = 2048 bytes per step
// (ISA 10.x async pseudocode adds INST_OFFSET on both sides), so a single
// (lds_addr, voffset, saddr) triple covers the whole stage. x is a read-once
// 176 MB stream -> non-temporal hint keeps it from churning L2.
#define ALD(OFF) \
    "global_load_async_to_lds_b64 %0, %1, %2 offset:" OFF " th:TH_LOAD_NT\n"

__device__ __forceinline__ void async_stage_load(uint32_t ldsa, int voff,
                                                 const float* xbase) {
    asm volatile(
        ALD("0")     ALD("2048")  ALD("4096")  ALD("6144")
        ALD("8192")  ALD("10240") ALD("12288") ALD("14336")
        ALD("16384") ALD("18432") ALD("20480") ALD("22528")
        ALD("24576") ALD("26624") ALD("28672") ALD("30720")
        :
        : "v"(ldsa), "v"(voff), "s"(xbase)
        : "memory");
}

__global__ __launch_bounds__(NTHREADS) void
fast_multi_ema_mixture_kernel(const float* __restrict__ x,
                              const float* __restrict__ logit_alpha,
                              const float* __restrict__ mix_logits,
                              float* __restrict__ out) {
    // Triple-buffered stage: 3 * 16 * 512 floats = 96 KB LDS (of 320 KB/WGP).
    __shared__ float smem[NBUF * TT * CC];

    const int tid = threadIdx.x;
    const int b   = blockIdx.x;
    const int c0  = tid * VEC;

    // ---------------- per-thread parameter precompute ----------------
    // a = clamp(sigmoid(logit_alpha)); mix = softmax_K(mix_logits[c,:])
    float a[VEC][KK], ia[VEC][KK], w[VEC][KK], ip[VEC][KK], S[VEC][KK], mw[VEC][KK];
#pragma unroll
    for (int v = 0; v < VEC; ++v) {
        const int c = c0 + v;
        float e[KK];
        float mmax = -1e30f;
#pragma unroll
        for (int k = 0; k < KK; ++k) {
            const float la = logit_alpha[k * CC + c];
            float av = 1.0f / (1.0f + __expf(-la));
            av = fminf(fmaxf(av, 1e-4f), 1.0f - 1e-4f);
            a[v][k]  = av;
            ia[v][k] = 1.0f / av;      // a^{-1}
            w[v][k]  = 1.0f - av;      // becomes (1-a)*a^t via w *= a
            ip[v][k] = 1.0f;           // a^{-t}, clamped to 1e8 (== 1/max(a^t,1e-8))
            S[v][k]  = 0.0f;           // running cumsum of x*weights
            const float ml = mix_logits[c * KK + k];
            e[k] = ml;
            mmax = fmaxf(mmax, ml);
        }
        float sum = 0.0f;
#pragma unroll
        for (int k = 0; k < KK; ++k) { e[k] = __expf(e[k] - mmax); sum += e[k]; }
        const float rs = 1.0f / sum;
#pragma unroll
        for (int k = 0; k < KK; ++k) mw[v][k] = e[k] * rs;
    }

    // ---------------- async triple-buffered pipeline ----------------
    const int base_elem = (b * TL) * CC + c0;               // t = 0 element index
    int voff = base_elem * 4;                               // byte offset into x
    uint32_t ldsa[NBUF];
#pragma unroll
    for (int p = 0; p < NBUF; ++p)
        ldsa[p] = (uint32_t)(uintptr_t)(&smem[p * TT * CC + c0]);

    // Prefetch stages 0 and 1 (32 ops in flight = 8 KB/wave; ~16 MB device-wide
    // against a ~12-23 MB bandwidth-delay product at 23.3 TB/s).
    async_stage_load(ldsa[0], voff, x);
    voff += TT * CC * 4;
    async_stage_load(ldsa[1], voff, x);
    voff += TT * CC * 4;

    float* __restrict__ outp = out + base_elem;

    for (int s = 0; s < NSTAGE; ++s) {
        const int pc = s % NBUF;
        const float* __restrict__ sbuf = smem + pc * TT * CC;

        if (s + 2 < NSTAGE) {
            async_stage_load(ldsa[(s + 2) % NBUF], voff, x);    // prefetch s+2
            voff += TT * CC * 4;
            asm volatile("s_wait_asynccnt 0x20" ::: "memory");  // stage s resident
        } else if (s + 1 < NSTAGE) {
            asm volatile("s_wait_asynccnt 0x10" ::: "memory");
        } else {
            asm volatile("s_wait_asynccnt 0x0" ::: "memory");
        }

        // Each thread consumes only the LDS bytes its own async ops wrote:
        // no block barrier needed anywhere.
#pragma unroll
        for (int j = 0; j < TT; ++j) {
            const int t = s * TT + j;
            const vfloat2 xv = *(const vfloat2*)(sbuf + j * CC + c0);
            const float xs[VEC] = {xv.x, xv.y};
            float o[VEC];
#pragma unroll
            for (int v = 0; v < VEC; ++v) {
                float acc = 0.0f;
#pragma unroll
                for (int k = 0; k < KK; ++k) {
                    if (j == 0 && s == 0) {
                        S[v][k] = xs[v];                     // t==0: weight is 1
                    } else {
                        w[v][k] *= a[v][k];                  // (1-a) * a^t
                        ip[v][k] = fminf(ip[v][k] * ia[v][k], 1e8f); // 1/max(a^t,1e-8)
                        S[v][k] = fmaf(xs[v], w[v][k], S[v][k]);
                    }
                    acc = fmaf(S[v][k] * ip[v][k], mw[v][k], acc);
                }
                o[v] = acc;
            }
            vfloat2 ov;
            ov.x = o[0];
            ov.y = o[1];
            // Write-once 176 MB stream: keep it out of L2 with NT stores.
            __builtin_nontemporal_store(ov, (vfloat2*)(outp + t * CC));
        }
    }
}

extern "C" void kernel_launch(void* const* d_in, const int* in_sizes, int n_in,
                              void* d_out, int out_size, void* d_ws, size_t ws_size,
                              hipStream_t stream) {
    (void)in_sizes; (void)n_in; (void)out_size; (void)d_ws; (void)ws_size;
    const float* x           = (const float*)d_in[0];  // [B, T, C]
    const float* logit_alpha = (const float*)d_in[1];  // [K, C]
    const float* mix_logits  = (const float*)d_in[2];  // [C, K]
    float* out               = (float*)d_out;          // [B, T, C]

    dim3 grid(BB);
    dim3 block(NTHREADS);
    hipLaunchKernelGGL(fast_multi_ema_mixture_kernel, grid, block, 0, stream,
                       x, logit_alpha, mix_logits, out);
}